// ImprovedGNNWithEmbeddings_34686155883084
// MI455X (gfx1250) — compile-verified
//
#include <hip/hip_runtime.h>
#include <hip/hip_bf16.h>

// ---------------------------------------------------------------------------
// GINE GNN forward on MI455X (gfx1250, wave32).
//   N=100000 nodes, E=1600000 edges, NODE_IN=6, EDGE_IN=4, HID=128, OUT=4, L=5
// Edge phase: L2-resident gather/scatter (x, agg each 51MB << 192MB L2),
//   one wave per edge, lane owns 4 columns -> coalesced 512B row bursts.
// GEMMs: full-precision f32 WMMA (V_WMMA_F32_16X16X4_F32), K templated so the
//   inner loop is branch-free; W staged through LDS in 128-row chunks with a
//   (k,k+1)-pair swizzle so each B fragment is one ds_load_b64;
//   bias/relu/BN-stats fused into epilogues.
// ---------------------------------------------------------------------------

typedef __attribute__((ext_vector_type(2))) float v2f;
typedef __attribute__((ext_vector_type(8))) float v8f;

#define HID 128

// ---------------------------------------------------------------------------
// Edge kernel, in_dim = 6 (layer 0): one thread per edge.
// ---------------------------------------------------------------------------
__global__ __launch_bounds__(256)
void edge_small_kernel(const float* __restrict__ x, const long long* __restrict__ ei,
                       const float* __restrict__ ea, const float* __restrict__ We,
                       const float* __restrict__ be, float* __restrict__ agg, int E)
{
    int e = blockIdx.x * blockDim.x + threadIdx.x;
    int stride = gridDim.x * blockDim.x;
    for (; e < E; e += stride) {
        long long s = ei[e];
        long long d = ei[(size_t)E + e];
        float4 av = *(const float4*)(ea + 4 * (size_t)e);
        const float* xs = x + (size_t)s * 6;
        float* ag = agg + (size_t)d * 6;
#pragma unroll
        for (int j = 0; j < 6; ++j) {
            float v = be[j] + av.x * We[0 * 6 + j] + av.y * We[1 * 6 + j]
                            + av.z * We[2 * 6 + j] + av.w * We[3 * 6 + j];
            v = fmaxf(xs[j] + v, 0.0f);
            atomicAdd(&ag[j], v);
        }
    }
}

// ---------------------------------------------------------------------------
// Edge kernel, in_dim = 128: one wave (32 lanes) per edge, lane owns 4 cols.
// x[src] row = one coalesced 512B load; We fragment hoisted out of edge loop.
// ---------------------------------------------------------------------------
__global__ __launch_bounds__(256)
void edge_big_kernel(const float* __restrict__ x, const long long* __restrict__ ei,
                     const float* __restrict__ ea, const float* __restrict__ We,
                     const float* __restrict__ be, float* __restrict__ agg, int E)
{
    int lane = threadIdx.x & 31;
    int wid  = (blockIdx.x * blockDim.x + threadIdx.x) >> 5;
    int nw   = (gridDim.x * blockDim.x) >> 5;
    int c0   = lane * 4;

    // Per-lane weight fragment: We[k][c0..c0+3], k=0..3, and bias fragment.
    const float4* We4 = (const float4*)We;
    float4 w0 = We4[0 * 32 + lane];
    float4 w1 = We4[1 * 32 + lane];
    float4 w2 = We4[2 * 32 + lane];
    float4 w3 = We4[3 * 32 + lane];
    float4 bb = ((const float4*)be)[lane];

    for (int e = wid; e < E; e += nw) {
        long long s = ei[e];
        long long d = ei[(size_t)E + e];
        // Prefetch next edge's streaming operands (global_prefetch_b8).
        int en = e + nw;
        if (en < E) {
            __builtin_prefetch(ea + 4 * (size_t)en, 0, 0);
            __builtin_prefetch(ei + en, 0, 0);
        }
        float4 av = *(const float4*)(ea + 4 * (size_t)e);
        float4 xs = *(const float4*)(x + (size_t)s * HID + c0);
        float4 m;
        m.x = fmaxf(xs.x + bb.x + av.x * w0.x + av.y * w1.x + av.z * w2.x + av.w * w3.x, 0.0f);
        m.y = fmaxf(xs.y + bb.y + av.x * w0.y + av.y * w1.y + av.z * w2.y + av.w * w3.y, 0.0f);
        m.z = fmaxf(xs.z + bb.z + av.x * w0.z + av.y * w1.z + av.z * w2.z + av.w * w3.z, 0.0f);
        m.w = fmaxf(xs.w + bb.w + av.x * w0.w + av.y * w1.w + av.z * w2.w + av.w * w3.w, 0.0f);
        float* ag = agg + (size_t)d * HID + c0;
        atomicAdd(ag + 0, m.x);
        atomicAdd(ag + 1, m.y);
        atomicAdd(ag + 2, m.z);
        atomicAdd(ag + 3, m.w);
    }
}

// ---------------------------------------------------------------------------
// WMMA GEMM: out[N,128] = act( (A [+ A2]) @ W[K,128] + bias ), optional BN
// stats. K is a compile-time constant -> branch-free inner loop, v2f A loads.
// Wave owns a 16-row strip x 128 cols (8 accumulator tiles).
//
// LDS pair-swizzle: for k-pair p (k = kc+2p, kc+2p+1) and column c,
//   ldsW[p*256 + 2c + 0] = W[kc+2p  ][c]
//   ldsW[p*256 + 2c + 1] = W[kc+2p+1][c]
// so a B fragment {W[k][col], W[k+1][col]} is ONE aligned ds_load_b64 with no
// register shuffles in the WMMA loop.
//
// f32 16x16x4 fragment layout (ISA 7.12.2):
//   A: lanes 0-15 rows M=0..15; vgpr0/1 = K 0/1 (lanes<16) or K 2/3 (lanes>=16)
//   B: vgpr0/1 = rows K 0/1 (lanes<16) or K 2/3; lane%16 = col N
//   D: vgpr j -> row j (lanes<16) / row j+8 (lanes>=16); lane%16 = col N
// ---------------------------------------------------------------------------
template <int K, int FUSE_ADD, int RELU, int STATS>
__global__ __launch_bounds__(256)
void gemm_wmma_kernel(const float* __restrict__ A, const float* __restrict__ A2,
                      const float* __restrict__ W, const float* __restrict__ bias,
                      float* __restrict__ out, float* __restrict__ stats, int N)
{
    constexpr int K4  = (K + 3) >> 2;          // k-steps of 4
    constexpr int KP4 = K4 * 4;                // zero-padded K
    constexpr int NCH = (KP4 + 127) >> 7;      // 128-row LDS chunks

    __shared__ float ldsW[HID * HID];          // 64 KB (of 320 KB/WGP)

    const int wave    = threadIdx.x >> 5;      // 0..7
    const int lane    = threadIdx.x & 31;
    const int halfSel = lane >> 4;             // selects K-pair and D row half
    const int ln      = lane & 15;

    const int nRowTiles = N >> 4;              // N % 16 == 0 (100000/16 = 6250)
    const int nGroups   = (nRowTiles + 7) >> 3;

    for (int grp = blockIdx.x; grp < nGroups; grp += gridDim.x) {
        const int  rowTile = grp * 8 + wave;
        const bool active  = rowTile < nRowTiles;
        const int  row0    = rowTile << 4;
        const int  arow    = row0 + ln;

        v8f acc[8];
#pragma unroll
        for (int ct = 0; ct < 8; ++ct) {
            float bv = bias[ct * 16 + ln];
#pragma unroll
            for (int j = 0; j < 8; ++j) acc[ct][j] = bv;
        }

#pragma unroll
        for (int ch = 0; ch < NCH; ++ch) {
            constexpr int ROWS_FULL = 128;
            const int kc   = ch * 128;
            const int rows = (KP4 - kc < ROWS_FULL) ? (KP4 - kc) : ROWS_FULL;
            const int pairs = rows >> 1;

            __syncthreads();
            // Stage W rows [kc, kc+rows) into LDS with the (k,k+1) pair
            // swizzle; float4-wide ds_store_b128, zero-padded beyond K.
            {
                float4* lds4 = (float4*)ldsW;
                const int nq = pairs * 64;     // float4s this chunk
                for (int q = threadIdx.x; q < nq; q += 256) {
                    int p  = q >> 6;           // k-pair within chunk
                    int c2 = q & 63;           // column pair
                    int r0 = kc + 2 * p;
                    v2f wa = {0.0f, 0.0f}, wb = {0.0f, 0.0f};
                    if (r0 < K)     wa = *(const v2f*)(W + (size_t)r0 * HID + 2 * c2);
                    if (r0 + 1 < K) wb = *(const v2f*)(W + (size_t)(r0 + 1) * HID + 2 * c2);
                    lds4[q] = make_float4(wa.x, wb.x, wa.y, wb.y);
                }
            }
            __syncthreads();

            if (active) {
                const int steps = rows >> 2;
#pragma unroll 4
                for (int kk = 0; kk < steps; ++kk) {
                    const int k0 = kc + kk * 4 + 2 * halfSel;  // global k
                    const int pl = kk * 2 + halfSel;           // LDS pair index
                    v2f a;
                    if (K & 3) {
                        // Tail-padded path (layer-0 K=6 only).
                        a.x = (k0     < K) ? A[(size_t)arow * K + k0]     : 0.0f;
                        a.y = (k0 + 1 < K) ? A[(size_t)arow * K + k0 + 1] : 0.0f;
                        if (FUSE_ADD) {
                            a.x += (k0     < K) ? A2[(size_t)arow * K + k0]     : 0.0f;
                            a.y += (k0 + 1 < K) ? A2[(size_t)arow * K + k0 + 1] : 0.0f;
                        }
                    } else {
                        a = *(const v2f*)(A + (size_t)arow * K + k0);      // b64 load
                        if (FUSE_ADD) {
                            v2f a2 = *(const v2f*)(A2 + (size_t)arow * K + k0);
                            a.x += a2.x; a.y += a2.y;
                        }
                    }
                    const float* lrow = ldsW + pl * 256;
#pragma unroll
                    for (int ct = 0; ct < 8; ++ct) {
                        const int col = ct * 16 + ln;
                        v2f b = *(const v2f*)(lrow + 2 * col);  // ds_load_b64
                        acc[ct] = __builtin_amdgcn_wmma_f32_16x16x4_f32(
                            false, a, false, b, (short)0, acc[ct], false, false);
                    }
                }
            }
        }

        if (active) {
#pragma unroll
            for (int ct = 0; ct < 8; ++ct) {
                const int col = ct * 16 + ln;
                float s = 0.0f, s2 = 0.0f;
#pragma unroll
                for (int j = 0; j < 8; ++j) {
                    float v = acc[ct][j];
                    if (RELU) v = fmaxf(v, 0.0f);
                    int row = row0 + j + 8 * halfSel;
                    out[(size_t)row * HID + col] = v;
                    if (STATS) { s += v; s2 += v * v; }
                }
                if (STATS) {
                    // lanes L and L+16 hold the same column -> fold (wave32)
                    s  += __shfl_down(s, 16, 32);
                    s2 += __shfl_down(s2, 16, 32);
                    if (halfSel == 0) {
                        atomicAdd(&stats[col], s);
                        atomicAdd(&stats[HID + col], s2);
                    }
                }
            }
        }
    }
}

// ---------------------------------------------------------------------------
// BN finalize: 128 threads -> per-channel scale/shift.
// ---------------------------------------------------------------------------
__global__ void bn_finalize_kernel(const float* __restrict__ stats,
                                   const float* __restrict__ gamma,
                                   const float* __restrict__ beta,
                                   float* __restrict__ scsh, float invN)
{
    int c = threadIdx.x;
    float mu  = stats[c] * invN;
    float var = stats[HID + c] * invN - mu * mu;
    float sc  = gamma[c] * rsqrtf(var + 1e-5f);
    scsh[c] = sc;
    scsh[HID + c] = beta[c] - mu * sc;
}

// ---------------------------------------------------------------------------
// BN apply + relu; write next-layer input and the JK-concat slice.
// ---------------------------------------------------------------------------
__global__ __launch_bounds__(256)
void bn_apply_kernel(const float* __restrict__ H, const float* __restrict__ scsh,
                     float* __restrict__ xnext, float* __restrict__ feats,
                     int N, int featOff)
{
    size_t idx = (size_t)blockIdx.x * blockDim.x + threadIdx.x;
    size_t total = (size_t)N * HID;
    size_t stride = (size_t)gridDim.x * blockDim.x;
    for (; idx < total; idx += stride) {
        int c = (int)(idx & (HID - 1));
        size_t row = idx >> 7;
        float y = fmaxf(H[idx] * scsh[c] + scsh[HID + c], 0.0f);
        xnext[idx] = y;
        feats[row * (HID * 5) + featOff + c] = y;
    }
}

// ---------------------------------------------------------------------------
// fc2: [N,128] @ [128,4] + b  (tiny; plain VALU).
// ---------------------------------------------------------------------------
__global__ __launch_bounds__(256)
void fc2_kernel(const float* __restrict__ P, const float* __restrict__ W,
                const float* __restrict__ b, float* __restrict__ out, int N)
{
    int i = blockIdx.x * blockDim.x + threadIdx.x;
    int stride = gridDim.x * blockDim.x;
    for (; i < N; i += stride) {
        float o0 = b[0], o1 = b[1], o2 = b[2], o3 = b[3];
        const float* p = P + (size_t)i * HID;
#pragma unroll 4
        for (int k = 0; k < HID; ++k) {
            float pv = p[k];
            o0 += pv * W[k * 4 + 0];
            o1 += pv * W[k * 4 + 1];
            o2 += pv * W[k * 4 + 2];
            o3 += pv * W[k * 4 + 3];
        }
        float* o = out + (size_t)i * 4;
        o[0] = o0; o[1] = o1; o[2] = o2; o[3] = o3;
    }
}

// ---------------------------------------------------------------------------
// Host launch.  Input order (setup_inputs flattened):
//   [0]=x  [1]=edge_index(i64)  [2]=edge_attr
//   [3+8l .. 3+8l+7] = layer l: We, be, W1, b1, W2, b2, gamma, beta  (l=0..4)
//   [43]=Wfc1 [44]=bfc1 [45]=Wfc2 [46]=bfc2
// ---------------------------------------------------------------------------
extern "C" void kernel_launch(void* const* d_in, const int* in_sizes, int n_in,
                              void* d_out, int out_size, void* d_ws, size_t ws_size,
                              hipStream_t stream)
{
    const int N = in_sizes[0] / 6;
    const int E = in_sizes[2] / 4;

    const float*     x  = (const float*)d_in[0];
    const long long* ei = (const long long*)d_in[1];
    const float*     ea = (const float*)d_in[2];
    const float* Wfc1 = (const float*)d_in[43];
    const float* bfc1 = (const float*)d_in[44];
    const float* Wfc2 = (const float*)d_in[45];
    const float* bfc2 = (const float*)d_in[46];

    // Workspace layout (floats)
    float* ws = (float*)d_ws;
    size_t nh = (size_t)N * HID;
    float* xcur  = ws;                       // [N,128]
    float* agg   = ws + nh;                  // [N,in_dim] per layer
    float* P     = ws + 2 * nh;              // [N,128] GEMM1 out / fc1 out
    float* Hbuf  = ws + 3 * nh;              // [N,128] GEMM2 out
    float* feats = ws + 4 * nh;              // [N,640] JK concat
    float* stats = feats + (size_t)N * HID * 5;  // [256]
    float* scsh  = stats + 256;                  // [256]

    const int nRowTiles = N >> 4;
    const int gemmGrid  = (nRowTiles + 7) >> 3;   // 128 rows per block
    const int edgeGrid  = 2048;
    const int ewGrid    = 4096;

    for (int l = 0; l < 5; ++l) {
        const float* We    = (const float*)d_in[3 + 8 * l + 0];
        const float* be    = (const float*)d_in[3 + 8 * l + 1];
        const float* W1    = (const float*)d_in[3 + 8 * l + 2];
        const float* b1    = (const float*)d_in[3 + 8 * l + 3];
        const float* W2    = (const float*)d_in[3 + 8 * l + 4];
        const float* b2    = (const float*)d_in[3 + 8 * l + 5];
        const float* gamma = (const float*)d_in[3 + 8 * l + 6];
        const float* beta  = (const float*)d_in[3 + 8 * l + 7];

        const int in_dim = (l == 0) ? 6 : HID;
        const float* xin = (l == 0) ? x : xcur;

        hipMemsetAsync(agg, 0, (size_t)N * in_dim * sizeof(float), stream);
        hipMemsetAsync(stats, 0, 256 * sizeof(float), stream);

        if (l == 0) {
            edge_small_kernel<<<edgeGrid, 256, 0, stream>>>(xin, ei, ea, We, be, agg, E);
            gemm_wmma_kernel<6, 1, 1, 0><<<gemmGrid, 256, 0, stream>>>(
                xin, agg, W1, b1, P, nullptr, N);
        } else {
            edge_big_kernel<<<edgeGrid, 256, 0, stream>>>(xin, ei, ea, We, be, agg, E);
            gemm_wmma_kernel<HID, 1, 1, 0><<<gemmGrid, 256, 0, stream>>>(
                xin, agg, W1, b1, P, nullptr, N);
        }

        // H = P @ W2 + b2, with BN sum/sumsq accumulation
        gemm_wmma_kernel<HID, 0, 0, 1><<<gemmGrid, 256, 0, stream>>>(
            P, nullptr, W2, b2, Hbuf, stats, N);

        bn_finalize_kernel<<<1, HID, 0, stream>>>(stats, gamma, beta, scsh, 1.0f / (float)N);
        bn_apply_kernel<<<ewGrid, 256, 0, stream>>>(Hbuf, scsh, xcur, feats, N, HID * l);
    }

    // fc1: P = relu(feats @ Wfc1 + bfc1), K = 640, 5 LDS chunks
    gemm_wmma_kernel<HID * 5, 0, 1, 0><<<gemmGrid, 256, 0, stream>>>(
        feats, nullptr, Wfc1, bfc1, P, nullptr, N);

    // fc2: out = P @ Wfc2 + bfc2
    fc2_kernel<<<ewGrid, 256, 0, stream>>>(P, Wfc2, bfc2, (float*)d_out, N);
}